// GATConv_37160057045565
// MI455X (gfx1250) — compile-verified
//
#include <hip/hip_runtime.h>
#include <math.h>

#define HH 4
#define FF 32
#define INF_ 128
#define EFF 32
#define HF 128
#define NEG_SLOPE 0.2f

typedef __attribute__((ext_vector_type(2))) float v2f;
typedef __attribute__((ext_vector_type(8))) float v8f;

// ---------------------------------------------------------------- init
__global__ void k_init(float* out, long out_n, float* denom, unsigned* emax,
                       float* deg, int n_nodes, int nh) {
    long i = (long)blockIdx.x * blockDim.x + threadIdx.x;
    long stride = (long)gridDim.x * blockDim.x;
    for (long j = i; j < out_n; j += stride) out[j] = 0.f;
    for (long j = i; j < nh; j += stride) { denom[j] = 0.f; emax[j] = 0u; }
    for (long j = i; j < n_nodes; j += stride) deg[j] = 0.f;
}

// ------------------------------------------------- fold fc_edge_w * attn_edge
__global__ void k_wee(const float* __restrict__ fc_edge_w,
                      const float* __restrict__ attn_edge,
                      float* __restrict__ wee) {
    int t = threadIdx.x;           // 0..127
    int h = t / EFF, ef = t % EFF;
    float s = 0.f;
    for (int f = 0; f < FF; ++f)
        s += attn_edge[h * FF + f] * fc_edge_w[(h * FF + f) * EFF + ef];
    wee[t] = s;
}

// --------------------------------------------- node GEMMs via fp32 WMMA
// C_fc = feat @ fc_w.T ; C_res = feat @ res_w.T   (both [N,128])
__global__ void __launch_bounds__(256)
k_node_gemm(const float* __restrict__ feat,
            const float* __restrict__ fc_w,
            const float* __restrict__ res_w,
            float* __restrict__ feat_src,
            float* __restrict__ resval, int n_nodes) {
    __shared__ float lds_a[16 * 132];      // 16 rows x 128 cols, pitch 132 (no bank conflicts)
    const int m0   = blockIdx.x * 16;
    const int t    = threadIdx.x;
    const int lane = t & 31;
    const int wave = t >> 5;               // 0..7  -> 16-col slice each

    // cooperative A-tile load (coalesced float4)
    {
        int row = t >> 4;                  // 0..15
        int c0  = (t & 15) * 8;            // 0..120
        int rowg = m0 + row; if (rowg >= n_nodes) rowg = n_nodes - 1;
        const float4* s4 = (const float4*)(feat + (size_t)rowg * INF_ + c0);
        float4 a0 = s4[0], a1 = s4[1];
        float* d = lds_a + row * 132 + c0;
        d[0]=a0.x; d[1]=a0.y; d[2]=a0.z; d[3]=a0.w;
        d[4]=a1.x; d[5]=a1.y; d[6]=a1.z; d[7]=a1.w;
    }
    __syncthreads();

    const int col   = wave * 16 + (lane & 15);   // output column (0..127)
    const int khalf = (lane >> 4) * 2;           // lanes 16-31 carry K+2,K+3
    const int arow  = lane & 15;

    v8f acc_fc  = {};
    v8f acc_res = {};
    for (int k = 0; k < INF_; k += 4) {
        v2f a, bf, br;
        a[0]  = lds_a[arow * 132 + k + khalf];
        a[1]  = lds_a[arow * 132 + k + khalf + 1];
        bf[0] = fc_w[(size_t)col * INF_ + k + khalf];
        bf[1] = fc_w[(size_t)col * INF_ + k + khalf + 1];
        br[0] = res_w[(size_t)col * INF_ + k + khalf];
        br[1] = res_w[(size_t)col * INF_ + k + khalf + 1];
        acc_fc  = __builtin_amdgcn_wmma_f32_16x16x4_f32(false, a, false, bf,
                                                        (short)0, acc_fc, false, false);
        acc_res = __builtin_amdgcn_wmma_f32_16x16x4_f32(false, a, false, br,
                                                        (short)0, acc_res, false, false);
    }

    const int mbase = m0 + ((lane >> 4) << 3);   // lanes<16 -> rows 0..7, else 8..15
#pragma unroll
    for (int r = 0; r < 8; ++r) {
        int m = mbase + r;
        if (m < n_nodes) {
            feat_src[(size_t)m * HF + col] = acc_fc[r];
            resval [(size_t)m * HF + col] = acc_res[r];
        }
    }
}

// ------------------------------------------------- per-node attention dots
__global__ void k_elr(const float* __restrict__ feat_src,
                      const float* __restrict__ attn_l,
                      const float* __restrict__ attn_r,
                      float* __restrict__ el, float* __restrict__ er, int nh) {
    int i = blockIdx.x * blockDim.x + threadIdx.x;
    if (i >= nh) return;
    int n = i / HH, h = i % HH;
    const float4* fs = (const float4*)(feat_src + (size_t)n * HF + h * FF);
    const float4* al = (const float4*)(attn_l + h * FF);
    const float4* ar = (const float4*)(attn_r + h * FF);
    float sl = 0.f, sr = 0.f;
#pragma unroll
    for (int j = 0; j < 8; ++j) {
        float4 v = fs[j], l = al[j], r = ar[j];
        sl += v.x*l.x + v.y*l.y + v.z*l.z + v.w*l.w;
        sr += v.x*r.x + v.y*r.y + v.z*r.z + v.w*r.w;
    }
    el[i] = sl; er[i] = sr;
}

// --------------------------------------------- ordered-uint float max trick
__device__ __forceinline__ unsigned f2ord(float f) {
    unsigned u = __float_as_uint(f);
    return (u & 0x80000000u) ? ~u : (u | 0x80000000u);
}
__device__ __forceinline__ float ord2f(unsigned k) {
    unsigned u = (k & 0x80000000u) ? (k & 0x7FFFFFFFu) : ~k;
    return __uint_as_float(u);
}

// ------------------------------------------------- edge scores + seg-max + deg
__global__ void k_edge_score(const float* __restrict__ edge_fea,
                             const int* __restrict__ src, const int* __restrict__ dst,
                             const float* __restrict__ wee,
                             const float* __restrict__ el, const float* __restrict__ er,
                             float* __restrict__ eval, unsigned* __restrict__ emax,
                             float* __restrict__ deg, int E) {
    int e = blockIdx.x * blockDim.x + threadIdx.x;
    if (e >= E) return;
    int s = src[e], d = dst[e];
    float ee[HH] = {0.f, 0.f, 0.f, 0.f};
    const float4* ef4 = (const float4*)(edge_fea + (size_t)e * EFF);
#pragma unroll
    for (int j = 0; j < 8; ++j) {
        float4 v = ef4[j];
#pragma unroll
        for (int h = 0; h < HH; ++h) {
            float4 w = ((const float4*)(wee + h * EFF))[j];
            ee[h] += v.x*w.x + v.y*w.y + v.z*w.z + v.w*w.w;
        }
    }
#pragma unroll
    for (int h = 0; h < HH; ++h) {
        float x = el[s * HH + h] + er[d * HH + h] + ee[h];
        x = (x > 0.f) ? x : NEG_SLOPE * x;               // leaky relu
        eval[(size_t)e * HH + h] = x;
        atomicMax(&emax[d * HH + h], f2ord(x));
    }
    atomicAdd(&deg[d], 1.0f);
}

// ------------------------------------------------- exp + seg-sum(denominator)
__global__ void k_edge_exp(const int* __restrict__ dst, float* __restrict__ eval,
                           const unsigned* __restrict__ emax,
                           float* __restrict__ denom, int E) {
    int e = blockIdx.x * blockDim.x + threadIdx.x;
    if (e >= E) return;
    int d = dst[e];
#pragma unroll
    for (int h = 0; h < HH; ++h) {
        float m = ord2f(emax[d * HH + h]);
        if (!(m > -1e38f && m < 1e38f)) m = 0.f;         // guard (matches isfinite)
        float v = expf(eval[(size_t)e * HH + h] - m);
        eval[(size_t)e * HH + h] = v;
        atomicAdd(&denom[d * HH + h], v);
    }
}

// ------------------------------------------------- message scatter (1 wave/edge)
__global__ void __launch_bounds__(256)
k_scatter(const int* __restrict__ src, const int* __restrict__ dst,
          const float* __restrict__ feat_src, const float* __restrict__ eval,
          const float* __restrict__ denom, float* __restrict__ rst, int E) {
    int lane = threadIdx.x & 31;
    int e = blockIdx.x * (blockDim.x >> 5) + (threadIdx.x >> 5);
    if (e >= E) return;
    int s = src[e], d = dst[e];
    int h = lane >> 3;                                   // (lane*4)/32
    float a = eval[(size_t)e * HH + h] / denom[(size_t)d * HH + h];
    float4 v = ((const float4*)(feat_src + (size_t)s * HF))[lane];
    float* o = rst + (size_t)d * HF + lane * 4;
    atomicAdd(o + 0, v.x * a);
    atomicAdd(o + 1, v.y * a);
    atomicAdd(o + 2, v.z * a);
    atomicAdd(o + 3, v.w * a);
}

// ------------------------------------------------- finalize + graph readout
__global__ void __launch_bounds__(128)
k_finalize(const float* __restrict__ resval, const float* __restrict__ deg,
           float* __restrict__ rst, float* __restrict__ graph, int n_nodes) {
    int lane = threadIdx.x & 31;                         // float4 of the 128 feats
    int sub  = threadIdx.x >> 5;                         // 4 node rows per block
    float ax = 0.f, ay = 0.f, az = 0.f, aw = 0.f;
    for (int n = blockIdx.x * 4 + sub; n < n_nodes; n += gridDim.x * 4) {
        float dg = deg[n]; if (dg < 1.f) dg = 1.f;
        float inv = 1.f / dg;
        float4 r  = ((float4*)(rst + (size_t)n * HF))[lane];
        float4 rv = ((const float4*)(resval + (size_t)n * HF))[lane];
        r.x = r.x * inv + rv.x; r.y = r.y * inv + rv.y;
        r.z = r.z * inv + rv.z; r.w = r.w * inv + rv.w;
        r.x = r.x > 0.f ? r.x : 0.f;  r.y = r.y > 0.f ? r.y : 0.f;
        r.z = r.z > 0.f ? r.z : 0.f;  r.w = r.w > 0.f ? r.w : 0.f;
        ((float4*)(rst + (size_t)n * HF))[lane] = r;
        ax += r.x; ay += r.y; az += r.z; aw += r.w;
    }
    __shared__ float red[4][128][4];
    red[sub][lane * 4 + 0][0] = ax;  // keep layout simple: [sub][feat4*4+c] unused dims folded
    red[sub][lane * 4 + 1][0] = ay;
    red[sub][lane * 4 + 2][0] = az;
    red[sub][lane * 4 + 3][0] = aw;
    __syncthreads();
    if (sub == 0) {
#pragma unroll
        for (int c = 0; c < 4; ++c) {
            int f = lane * 4 + c;
            float sum = red[0][f][0] + red[1][f][0] + red[2][f][0] + red[3][f][0];
            atomicAdd(graph + f, sum);
        }
    }
}

// ----------------------------------------------------------------- launch
extern "C" void kernel_launch(void* const* d_in, const int* in_sizes, int n_in,
                              void* d_out, int out_size, void* d_ws, size_t ws_size,
                              hipStream_t stream) {
    const float* feat      = (const float*)d_in[0];
    const float* edge_fea  = (const float*)d_in[1];
    const int*   src       = (const int*)  d_in[2];
    const int*   dst       = (const int*)  d_in[3];
    const float* fc_w      = (const float*)d_in[4];
    const float* fc_edge_w = (const float*)d_in[5];
    const float* attn_l    = (const float*)d_in[6];
    const float* attn_r    = (const float*)d_in[7];
    const float* attn_edge = (const float*)d_in[8];
    const float* res_w     = (const float*)d_in[9];

    const int n = in_sizes[0] / INF_;
    const int E = in_sizes[2];
    const int nh = n * HH;

    float* graph = (float*)d_out;        // [H*F]
    float* rst   = graph + HF;           // [n*H*F]

    char* w = (char*)d_ws;
    float*    ws_feat_src = (float*)w;    w += (size_t)n * HF * 4;
    float*    ws_resval   = (float*)w;    w += (size_t)n * HF * 4;
    float*    ws_el       = (float*)w;    w += (size_t)nh * 4;
    float*    ws_er       = (float*)w;    w += (size_t)nh * 4;
    float*    ws_eval     = (float*)w;    w += (size_t)E * HH * 4;
    unsigned* ws_emax     = (unsigned*)w; w += (size_t)nh * 4;
    float*    ws_denom    = (float*)w;    w += (size_t)nh * 4;
    float*    ws_wee      = (float*)w;    w += (size_t)HH * EFF * 4;
    float*    ws_deg      = (float*)w;    w += (size_t)n * 4;

    long out_n = (long)HF + (long)n * HF;

    k_init<<<4096, 256, 0, stream>>>(graph, out_n, ws_denom, ws_emax, ws_deg, n, nh);
    k_wee<<<1, 128, 0, stream>>>(fc_edge_w, attn_edge, ws_wee);
    k_node_gemm<<<(n + 15) / 16, 256, 0, stream>>>(feat, fc_w, res_w,
                                                   ws_feat_src, ws_resval, n);
    k_elr<<<(nh + 255) / 256, 256, 0, stream>>>(ws_feat_src, attn_l, attn_r,
                                                ws_el, ws_er, nh);
    k_edge_score<<<(E + 255) / 256, 256, 0, stream>>>(edge_fea, src, dst, ws_wee,
                                                      ws_el, ws_er, ws_eval,
                                                      ws_emax, ws_deg, E);
    k_edge_exp<<<(E + 255) / 256, 256, 0, stream>>>(dst, ws_eval, ws_emax,
                                                    ws_denom, E);
    k_scatter<<<(E + 7) / 8, 256, 0, stream>>>(src, dst, ws_feat_src, ws_eval,
                                               ws_denom, rst, E);
    k_finalize<<<512, 128, 0, stream>>>(ws_resval, ws_deg, rst, graph, n);
}